// SpliceAI_V_DCN_62637803045634
// MI455X (gfx1250) — compile-verified
//
#include <hip/hip_runtime.h>

typedef __attribute__((ext_vector_type(2))) float v2f;
typedef __attribute__((ext_vector_type(8))) float v8f;

#define CCH 16
#define NB 4
#define LLEN 15000
#define LOUT 5000
#define KMAX 41

// ---------------------------------------------------------------------------
// input conv: x (B,L,4) -> conv (B,16,L), 1x1 over 4 input channels
// ---------------------------------------------------------------------------
__global__ void input_conv_kernel(const float* __restrict__ x,
                                  const float* __restrict__ w,   // (16,4)
                                  const float* __restrict__ bias,
                                  float* __restrict__ out) {
  int idx = blockIdx.x * blockDim.x + threadIdx.x;           // (b,o,l)
  if (idx >= NB * CCH * LLEN) return;
  int l = idx % LLEN;
  int o = (idx / LLEN) % CCH;
  int b = idx / (LLEN * CCH);
  const float* xp = x + (b * LLEN + l) * 4;
  float s = bias[o];
  s += w[o * 4 + 0] * xp[0];
  s += w[o * 4 + 1] * xp[1];
  s += w[o * 4 + 2] * xp[2];
  s += w[o * 4 + 3] * xp[3];
  out[idx] = s;
}

// ---------------------------------------------------------------------------
// BN stats: one block per channel, reduce over (B,L). stats[c]=mean,
// stats[16+c]=rsqrt(var+eps)
// ---------------------------------------------------------------------------
__global__ void bn_stats_kernel(const float* __restrict__ x,
                                float* __restrict__ stats) {
  int c = blockIdx.x;
  float s = 0.f, s2 = 0.f;
  for (int i = threadIdx.x; i < NB * LLEN; i += blockDim.x) {
    int b = i / LLEN, l = i % LLEN;
    float v = x[(b * CCH + c) * LLEN + l];
    s += v;
    s2 += v * v;
  }
  __shared__ float sh[256], sh2[256];
  sh[threadIdx.x] = s;
  sh2[threadIdx.x] = s2;
  __syncthreads();
  for (int ofs = 128; ofs > 0; ofs >>= 1) {
    if ((int)threadIdx.x < ofs) {
      sh[threadIdx.x] += sh[threadIdx.x + ofs];
      sh2[threadIdx.x] += sh2[threadIdx.x + ofs];
    }
    __syncthreads();
  }
  if (threadIdx.x == 0) {
    float N = (float)(NB * LLEN);
    float m = sh[0] / N;
    float var = sh2[0] / N - m * m;
    stats[c] = m;
    stats[CCH + c] = rsqrtf(var + 1e-3f);
  }
}

// ---------------------------------------------------------------------------
// BN apply + ReLU (elementwise)
// ---------------------------------------------------------------------------
__global__ void bn_relu_kernel(const float* __restrict__ x,
                               const float* __restrict__ g,
                               const float* __restrict__ bb,
                               const float* __restrict__ stats,
                               float* __restrict__ out) {
  int idx = blockIdx.x * blockDim.x + threadIdx.x;
  if (idx >= NB * CCH * LLEN) return;
  int c = (idx / LLEN) % CCH;
  float v = (x[idx] - stats[c]) * stats[CCH + c] * g[c] + bb[c];
  out[idx] = v > 0.f ? v : 0.f;
}

// ---------------------------------------------------------------------------
// offset conv1d via WMMA: a (B,16,L) * w (K,16,K) -> off (B,L,K)
// M = offset output channel kk (padded to 16-row tiles), N = 16 positions,
// K-reduction = (tap t, 4-channel chunk) -> 4K wmma f32 16x16x4 ops.
// Rows kk >= K use clamped weight rows; their stores are masked off.
// ---------------------------------------------------------------------------
__global__ void offset_conv_wmma_kernel(const float* __restrict__ a,
                                        const float* __restrict__ w,
                                        const float* __restrict__ bias,
                                        float* __restrict__ off, int K) {
  const int tilesPerBatch = (LLEN + 15) >> 4;
  int nRowTiles = (K + 15) >> 4;
  int wave = (blockIdx.x * blockDim.x + threadIdx.x) >> 5;
  int lane = threadIdx.x & 31;
  if (wave >= NB * tilesPerBatch * nRowTiles) return;   // uniform per wave
  int rt = wave % nRowTiles;
  int tmp = wave / nRowTiles;
  int tile = tmp % tilesPerBatch;
  int b = tmp / tilesPerBatch;
  int n = lane & 15;
  int hi = (lane >= 16) ? 1 : 0;
  int l = (tile << 4) + n;
  int pad = (K - 1) / 2;
  int kkRow = rt * 16 + n;
  int kkW = kkRow < K ? kkRow : K - 1;                  // clamp A rows
  const float* abase = a + b * CCH * LLEN;
  __builtin_prefetch(w, 0, 3);                          // global_prefetch_b8
  v8f acc = {};
  for (int t = 0; t < K; ++t) {
    int li = l + t - pad;
    float m = (li >= 0 && li < LLEN) ? 1.0f : 0.0f;
    int lic = min(max(li, 0), LLEN - 1);
#pragma unroll
    for (int c0 = 0; c0 < CCH; c0 += 4) {
      int cA = c0 + (hi ? 2 : 0);
      int cB = cA + 1;
      v2f bm, am;
      bm.x = abase[cA * LLEN + lic] * m;
      bm.y = abase[cB * LLEN + lic] * m;
      am.x = w[(kkW * CCH + cA) * K + t];
      am.y = w[(kkW * CCH + cB) * K + t];
      acc = __builtin_amdgcn_wmma_f32_16x16x4_f32(false, am, false, bm,
                                                  (short)0, acc, false, false);
    }
  }
  if (l < LLEN) {
    int ob = hi ? 8 : 0;
#pragma unroll
    for (int r = 0; r < 8; ++r) {
      int kk = rt * 16 + ob + r;
      if (kk < K) off[(b * LLEN + l) * K + kk] = acc[r] + bias[kk];
    }
  }
}

// ---------------------------------------------------------------------------
// deformable conv, WMMA f32 16x16x4. One wave -> one (batch, 16-position)
// output tile of all 16 output channels.
//   D(16x16: o x pos) += A(16x4: o x {tap,4 chans}) * B(4x{16 pos})
// A layout: lane holds row M=lane&15; v.x = Klocal (hi?2:0), v.y = +1
// B layout: v.x = row Klocal (hi?2:0) at col N=lane&15, v.y = +1
// ---------------------------------------------------------------------------
__global__ void deform_wmma_kernel(const float* __restrict__ a,
                                   const float* __restrict__ off,
                                   const float* __restrict__ w,     // (16,16,K)
                                   const float* __restrict__ bias,
                                   const float* __restrict__ residual, // or null
                                   float* __restrict__ out,
                                   int K, int dil) {
  const int tilesPerBatch = (LLEN + 15) >> 4;
  int wave = (blockIdx.x * blockDim.x + threadIdx.x) >> 5;
  int lane = threadIdx.x & 31;
  int b = wave / tilesPerBatch;
  if (b >= NB) return;                       // uniform per wave
  int tile = wave % tilesPerBatch;
  int n = lane & 15;
  int hi = (lane >= 16) ? 1 : 0;
  int l = (tile << 4) + n;
  int lc = l < LLEN ? l : LLEN - 1;
  int pad = dil * (K - 1) / 2;

  const float* abase = a + b * CCH * LLEN;
  const float* offb = off + (b * LLEN + lc) * K;
  __builtin_prefetch(w, 0, 3);               // global_prefetch_b8
  v8f acc = {};

  for (int t = 0; t < K; ++t) {
    float ofv = offb[t];
    float pos = (float)l + (float)(t * dil - pad) + ofv;
    float p0f = floorf(pos);
    float w1 = pos - p0f;
    float w0 = 1.0f - w1;
    int p0 = (int)p0f;
    int p1 = p0 + 1;
    float m0 = (p0 >= 0 && p0 < LLEN) ? 1.0f : 0.0f;
    float m1 = (p1 >= 0 && p1 < LLEN) ? 1.0f : 0.0f;
    int p0c = min(max(p0, 0), LLEN - 1);
    int p1c = min(max(p1, 0), LLEN - 1);
#pragma unroll
    for (int c0 = 0; c0 < CCH; c0 += 4) {
      int cA = c0 + (hi ? 2 : 0);
      int cB = cA + 1;
      const float* pa = abase + cA * LLEN;
      const float* pb = abase + cB * LLEN;
      v2f bm, am;
      bm.x = (pa[p0c] * m0) * w0 + (pa[p1c] * m1) * w1;   // samp(cA, l, t)
      bm.y = (pb[p0c] * m0) * w0 + (pb[p1c] * m1) * w1;   // samp(cB, l, t)
      am.x = w[(n * CCH + cA) * K + t];                   // w(o=n, cA, t)
      am.y = w[(n * CCH + cB) * K + t];
      acc = __builtin_amdgcn_wmma_f32_16x16x4_f32(false, am, false, bm,
                                                  (short)0, acc, false, false);
    }
  }

  if (l < LLEN) {
    int ob = hi ? 8 : 0;
#pragma unroll
    for (int r = 0; r < 8; ++r) {
      int oc = ob + r;
      float v = acc[r] + bias[oc];
      if (residual) v += residual[(b * CCH + oc) * LLEN + l];
      out[(b * CCH + oc) * LLEN + l] = v;
    }
  }
}

// ---------------------------------------------------------------------------
// 1x1 conv 16->16 via WMMA (K=16 as 4 chained 16x16x4), optional accumulate
// ---------------------------------------------------------------------------
__global__ void conv1x1_wmma_kernel(const float* __restrict__ in,
                                    const float* __restrict__ w,    // (16,16)
                                    const float* __restrict__ bias,
                                    float* __restrict__ out,
                                    int accumulate) {
  const int tilesPerBatch = (LLEN + 15) >> 4;
  int wave = (blockIdx.x * blockDim.x + threadIdx.x) >> 5;
  int lane = threadIdx.x & 31;
  int b = wave / tilesPerBatch;
  if (b >= NB) return;
  int tile = wave % tilesPerBatch;
  int n = lane & 15;
  int hi = (lane >= 16) ? 1 : 0;
  int l = (tile << 4) + n;
  int lc = l < LLEN ? l : LLEN - 1;
  const float* ibase = in + b * CCH * LLEN;
  v8f acc = {};
#pragma unroll
  for (int c0 = 0; c0 < CCH; c0 += 4) {
    int cA = c0 + (hi ? 2 : 0);
    int cB = cA + 1;
    v2f bm, am;
    bm.x = ibase[cA * LLEN + lc];
    bm.y = ibase[cB * LLEN + lc];
    am.x = w[n * CCH + cA];
    am.y = w[n * CCH + cB];
    acc = __builtin_amdgcn_wmma_f32_16x16x4_f32(false, am, false, bm,
                                                (short)0, acc, false, false);
  }
  if (l < LLEN) {
    int ob = hi ? 8 : 0;
#pragma unroll
    for (int r = 0; r < 8; ++r) {
      int oc = ob + r;
      float v = acc[r] + bias[oc];
      int di = (b * CCH + oc) * LLEN + l;
      out[di] = accumulate ? (out[di] + v) : v;
    }
  }
}

// ---------------------------------------------------------------------------
// final: crop skip middle LOUT and 1x1 conv 16->3 -> logits (B,3,LOUT)
// ---------------------------------------------------------------------------
__global__ void class_conv_kernel(const float* __restrict__ skip,
                                  const float* __restrict__ w,   // (3,16)
                                  const float* __restrict__ bias,
                                  float* __restrict__ out) {
  int idx = blockIdx.x * blockDim.x + threadIdx.x;   // (b,j,i)
  if (idx >= NB * 3 * LOUT) return;
  int i = idx % LOUT;
  int j = (idx / LOUT) % 3;
  int b = idx / (LOUT * 3);
  float s = bias[j];
  for (int c = 0; c < CCH; ++c)
    s += w[j * CCH + c] * skip[(b * CCH + c) * LLEN + (LLEN - LOUT) / 2 + i];
  out[idx] = s;
}

// ---------------------------------------------------------------------------
extern "C" void kernel_launch(void* const* d_in, const int* in_sizes, int n_in,
                              void* d_out, int out_size, void* d_ws, size_t ws_size,
                              hipStream_t stream) {
  (void)in_sizes; (void)n_in; (void)out_size; (void)ws_size;
  static const int KS[16] = {11,11,11,11,11,11,11,11,21,21,21,21,41,41,41,41};
  static const int DR[16] = {1,1,1,1,4,4,4,4,10,10,10,10,25,25,25,25};

  const float* x       = (const float*)d_in[0];
  const float* conv_w  = (const float*)d_in[1];
  const float* conv_b  = (const float*)d_in[2];
  const float* skip_w  = (const float*)d_in[3];
  const float* skip_b  = (const float*)d_in[4];
  const float* class_w = (const float*)d_in[5];
  const float* class_b = (const float*)d_in[6];
  // units: 12 tensors each starting at index 7; dense: 2 tensors each at 199
  const int UBASE = 7, DBASE = 7 + 16 * 12;

  float* ws    = (float*)d_ws;
  float* conv  = ws;                                  // B*16*L
  float* skip  = conv + NB * CCH * LLEN;
  float* abuf  = skip + NB * CCH * LLEN;
  float* dbuf  = abuf + NB * CCH * LLEN;
  float* offb  = dbuf + NB * CCH * LLEN;              // B*L*KMAX
  float* stats = offb + NB * LLEN * KMAX;             // 32 floats

  const int ELEMS = NB * CCH * LLEN;
  const int EBLK = (ELEMS + 255) / 256;
  const int tilesPerBatch = (LLEN + 15) / 16;
  const int waves = NB * tilesPerBatch;
  const int WBLK = (waves * 32 + 255) / 256;

  // input conv + skip init
  input_conv_kernel<<<EBLK, 256, 0, stream>>>(x, conv_w, conv_b, conv);
  conv1x1_wmma_kernel<<<WBLK, 256, 0, stream>>>(conv, skip_w, skip_b, skip, 0);

  int di = 0;
  for (int i = 0; i < 16; ++i) {
    int K = KS[i], D = DR[i];
    const float* const* u = (const float* const*)&d_in[UBASE + i * 12];
    const float* bn1_g = (const float*)u[0];
    const float* bn1_b = (const float*)u[1];
    const float* off1_w = (const float*)u[2];
    const float* off1_b = (const float*)u[3];
    const float* dc1_w = (const float*)u[4];
    const float* dc1_b = (const float*)u[5];
    const float* bn2_g = (const float*)u[6];
    const float* bn2_b = (const float*)u[7];
    const float* off2_w = (const float*)u[8];
    const float* off2_b = (const float*)u[9];
    const float* dc2_w = (const float*)u[10];
    const float* dc2_b = (const float*)u[11];

    int nRowTiles = (K + 15) / 16;
    int OBLK = (NB * tilesPerBatch * nRowTiles * 32 + 255) / 256;

    // stage 1: bn(conv) -> relu -> offsets -> deform -> dbuf
    bn_stats_kernel<<<CCH, 256, 0, stream>>>(conv, stats);
    bn_relu_kernel<<<EBLK, 256, 0, stream>>>(conv, bn1_g, bn1_b, stats, abuf);
    offset_conv_wmma_kernel<<<OBLK, 256, 0, stream>>>(abuf, off1_w, off1_b, offb, K);
    deform_wmma_kernel<<<WBLK, 256, 0, stream>>>(abuf, offb, dc1_w, dc1_b,
                                                 nullptr, dbuf, K, D);
    // stage 2: bn(dbuf) -> relu -> offsets -> deform (+residual conv) -> conv
    bn_stats_kernel<<<CCH, 256, 0, stream>>>(dbuf, stats);
    bn_relu_kernel<<<EBLK, 256, 0, stream>>>(dbuf, bn2_g, bn2_b, stats, abuf);
    offset_conv_wmma_kernel<<<OBLK, 256, 0, stream>>>(abuf, off2_w, off2_b, offb, K);
    deform_wmma_kernel<<<WBLK, 256, 0, stream>>>(abuf, offb, dc2_w, dc2_b,
                                                 conv, conv, K, D);

    if ((i + 1) % 4 == 0) {
      const float* dw = (const float*)d_in[DBASE + 2 * di];
      const float* db = (const float*)d_in[DBASE + 2 * di + 1];
      conv1x1_wmma_kernel<<<WBLK, 256, 0, stream>>>(conv, dw, db, skip, 1);
      ++di;
    }
  }

  int FBLK = (NB * 3 * LOUT + 255) / 256;
  class_conv_kernel<<<FBLK, 256, 0, stream>>>(skip, class_w, class_b,
                                              (float*)d_out);
}